// RMSAttn_55740085567782
// MI455X (gfx1250) — compile-verified
//
#include <hip/hip_runtime.h>

typedef float v2f __attribute__((ext_vector_type(2)));
typedef float v4f __attribute__((ext_vector_type(4)));
typedef float v8f __attribute__((ext_vector_type(8)));

__device__ __forceinline__ v2f mk2(float x, float y) { v2f r; r.x = x; r.y = y; return r; }

__device__ __forceinline__ v8f wmma4(v2f a, v2f b, v8f c) {
  // D = A(16x4 f32) * B(4x16 f32) + C(16x16 f32)
  return __builtin_amdgcn_wmma_f32_16x16x4_f32(false, a, false, b, (short)0, c, false, false);
}

// ---------------- constants ----------------
#define MM 16
#define HH 32
#define DD 128
#define PP 8192
#define NN 4096           // HH*DD
#define NTILES 513        // (PP+MM)/16
#define TILES_PER_CHUNK 9
#define NCHUNK 57         // 57*9 = 513

// ---------------- kernel 1: RMSNorm ----------------
__global__ void k_rmsnorm(const float* __restrict__ X, float* __restrict__ xn) {
  __shared__ float red[256];
  const int row = blockIdx.x;
  const float* x = X + (size_t)row * NN;
  float s = 0.f;
  for (int i = threadIdx.x; i < NN; i += 256) { float v = x[i]; s += v * v; }
  red[threadIdx.x] = s;
  __syncthreads();
  for (int off = 128; off > 0; off >>= 1) {
    if ((int)threadIdx.x < off) red[threadIdx.x] += red[threadIdx.x + off];
    __syncthreads();
  }
  const float scale = rsqrtf(red[0] / (float)NN);
  float* o = xn + (size_t)row * NN;
  for (int i = threadIdx.x; i < NN; i += 256) o[i] = x[i] * scale;
}

// ---------------- kernel 2: QKV projections (WMMA f32) ----------------
// out[m][j] = sum_k xn[m][k] * W[j][k]   (x @ W.T)
// grid = (NN/16, 3), block = 32 (one wave per 16x16 output tile)
__global__ void k_qkv(const float* __restrict__ xn,
                      const float* __restrict__ Wq,
                      const float* __restrict__ Wk,
                      const float* __restrict__ Wv,
                      float* __restrict__ q, float* __restrict__ k, float* __restrict__ v) {
  const int lane = threadIdx.x;
  const int jbase = blockIdx.x * 16;
  const float* W = (blockIdx.y == 0) ? Wq : (blockIdx.y == 1 ? Wk : Wv);
  float* out = (blockIdx.y == 0) ? q : (blockIdx.y == 1 ? k : v);

  const int half = lane >> 4;       // 0: K=0,1 slots  1: K=2,3 slots
  const int ln   = lane & 15;
  const int co   = half * 4;        // column sub-offset inside group-of-8

  const float* pA = xn + (size_t)ln * NN + co;                // A row = m
  const float* pB = W + (size_t)(jbase + ln) * NN + co;       // B col n -> W row jbase+n

  v8f acc0 = {}; v8f acc1 = {};
  for (int c = 0; c < NN; c += 16) {
    v4f a0 = *(const v4f*)(pA + c);
    v4f b0 = *(const v4f*)(pB + c);
    v4f a1 = *(const v4f*)(pA + c + 8);
    v4f b1 = *(const v4f*)(pB + c + 8);
    acc0 = wmma4(mk2(a0.x, a0.y), mk2(b0.x, b0.y), acc0);
    acc0 = wmma4(mk2(a0.z, a0.w), mk2(b0.z, b0.w), acc0);
    acc1 = wmma4(mk2(a1.x, a1.y), mk2(b1.x, b1.y), acc1);
    acc1 = wmma4(mk2(a1.z, a1.w), mk2(b1.z, b1.w), acc1);
  }
  v8f acc = acc0 + acc1;
  const int mbase = half * 8;
  float* orow = out + jbase + ln;
  #pragma unroll
  for (int r = 0; r < 8; ++r)
    orow[(size_t)(mbase + r) * NN] = acc[r];
}

// ---------------- kernel 3: zero accumulators ----------------
__global__ void k_zero(float* __restrict__ p, int n) {
  int i = blockIdx.x * 256 + threadIdx.x;
  if (i < n) p[i] = 0.f;
}

// ---------------- kernel 4: attention partials ----------------
// grid = (NCHUNK, HH), block = 32 (one wave).  Naive softmax is linear in
// partials: num[h][m][d] += sum_p exp(s) * V, den[h][m] += sum_p exp(s).
__global__ void k_attn(const float* __restrict__ q,
                       const float* __restrict__ kn,
                       const float* __restrict__ vn,
                       const float* __restrict__ cK,
                       const float* __restrict__ cV,
                       float* __restrict__ num, float* __restrict__ den) {
  __shared__ float wl[16 * 16];
  const int lane  = threadIdx.x;
  const int h     = blockIdx.y;
  const int chunk = blockIdx.x;
  const int half = lane >> 4;
  const int ln   = lane & 15;
  const int co   = half * 4;
  const int roff = half * 4;

  const float* pQ = q + (size_t)ln * NN + h * DD + co;        // A row = m
  const float* Kh = cK + (size_t)h * PP * DD;
  const float* Vh = cV + (size_t)h * PP * DD;

  v8f zero = {};
  v8f accO[8];
  #pragma unroll
  for (int t = 0; t < 8; ++t) accO[t] = zero;
  v8f accD = zero;
  const v2f ones = mk2(1.f, 1.f);

  for (int ti = 0; ti < TILES_PER_CHUNK; ++ti) {
    const int tile = chunk * TILES_PER_CHUNK + ti;   // 0..512
    const int pb = tile * 16;
    const bool newtile = (tile == 512);              // appended K/V rows

    // ---- scores S[m][n] = sum_d Q[m][d] * K[pb+n][d] ----
    const float* pK = newtile ? (kn + (size_t)ln * NN + h * DD + co)
                              : (Kh + (size_t)(pb + ln) * DD + co);
    v8f s = zero;
    #pragma unroll
    for (int c = 0; c < DD; c += 8) {
      v4f a = *(const v4f*)(pQ + c);
      v4f b = *(const v4f*)(pK + c);
      s = wmma4(mk2(a.x, a.y), mk2(b.x, b.y), s);
      s = wmma4(mk2(a.z, a.w), mk2(b.z, b.w), s);
    }

    // ---- exp, stage weight tile to LDS (C layout -> rows) ----
    {
      const int row0 = half * 8;
      #pragma unroll
      for (int r = 0; r < 8; ++r)
        wl[(row0 + r) * 16 + ln] = expf(s[r]);
    }
    __syncthreads();

    // re-load weights in A layout: lane m<16 gets cols c..c+3, lane m+16 cols c+4..c+7
    v4f aw0 = *(const v4f*)(&wl[ln * 16 + 0 + co]);
    v4f aw1 = *(const v4f*)(&wl[ln * 16 + 8 + co]);
    __syncthreads();

    // ---- row sums via ones-matrix WMMA ----
    accD = wmma4(mk2(aw0.x, aw0.y), ones, accD);
    accD = wmma4(mk2(aw0.z, aw0.w), ones, accD);
    accD = wmma4(mk2(aw1.x, aw1.y), ones, accD);
    accD = wmma4(mk2(aw1.z, aw1.w), ones, accD);

    // ---- O[m][n] += sum_kp W[m][kp] * V[pb+kp][dbase+n] ----
    const size_t rs = newtile ? (size_t)NN : (size_t)DD;
    const float* Vb = newtile ? (vn + h * DD + ln)
                              : (Vh + (size_t)pb * DD + ln);
    #pragma unroll
    for (int dt = 0; dt < 8; ++dt) {
      const float* Vc = Vb + dt * 16;
      // group c = 0
      {
        float b0 = Vc[(0 + roff) * rs];
        float b1 = Vc[(1 + roff) * rs];
        float b2 = Vc[(2 + roff) * rs];
        float b3 = Vc[(3 + roff) * rs];
        accO[dt] = wmma4(mk2(aw0.x, aw0.y), mk2(b0, b1), accO[dt]);
        accO[dt] = wmma4(mk2(aw0.z, aw0.w), mk2(b2, b3), accO[dt]);
      }
      // group c = 8
      {
        float b0 = Vc[(8 + roff) * rs];
        float b1 = Vc[(9 + roff) * rs];
        float b2 = Vc[(10 + roff) * rs];
        float b3 = Vc[(11 + roff) * rs];
        accO[dt] = wmma4(mk2(aw1.x, aw1.y), mk2(b0, b1), accO[dt]);
        accO[dt] = wmma4(mk2(aw1.z, aw1.w), mk2(b2, b3), accO[dt]);
      }
    }
  }

  // ---- combine partials ----
  const int mb = half * 8;
  #pragma unroll
  for (int dt = 0; dt < 8; ++dt) {
    #pragma unroll
    for (int r = 0; r < 8; ++r)
      atomicAdd(&num[((size_t)h * 16 + mb + r) * DD + dt * 16 + ln], accO[dt][r]);
  }
  if (ln == 0) {
    #pragma unroll
    for (int r = 0; r < 8; ++r)
      atomicAdd(&den[h * 16 + mb + r], accD[r]);
  }
}

// ---------------- kernel 5: final divide + layout ----------------
__global__ void k_final(const float* __restrict__ num, const float* __restrict__ den,
                        float* __restrict__ out) {
  int i = blockIdx.x * 256 + threadIdx.x;   // over 16*4096
  if (i >= MM * NN) return;
  int m = i / NN;
  int c = i % NN;
  int h = c / DD;
  int d = c % DD;
  out[i] = num[((size_t)h * 16 + m) * DD + d] / den[h * 16 + m];
}

extern "C" void kernel_launch(void* const* d_in, const int* in_sizes, int n_in,
                              void* d_out, int out_size, void* d_ws, size_t ws_size,
                              hipStream_t stream) {
  const float* X  = (const float*)d_in[0];
  const float* Wq = (const float*)d_in[1];
  const float* Wk = (const float*)d_in[2];
  const float* Wv = (const float*)d_in[3];
  const float* cK = (const float*)d_in[4];
  const float* cV = (const float*)d_in[5];

  float* ws  = (float*)d_ws;
  float* xn  = ws;                 // 16*4096
  float* q   = ws + 65536;         // 16*4096
  float* k   = ws + 131072;        // 16*4096
  float* v   = ws + 196608;        // 16*4096
  float* num = ws + 262144;        // 32*16*128 = 65536
  float* den = ws + 327680;        // 32*16     = 512

  k_rmsnorm<<<MM, 256, 0, stream>>>(X, xn);
  k_qkv<<<dim3(NN / 16, 3), 32, 0, stream>>>(xn, Wq, Wk, Wv, q, k, v);
  k_zero<<<(65536 + 512 + 255) / 256, 256, 0, stream>>>(num, 65536 + 512);
  k_attn<<<dim3(NCHUNK, HH), 32, 0, stream>>>(q, k, v, cK, cV, num, den);
  k_final<<<(MM * NN + 255) / 256, 256, 0, stream>>>(num, den, (float*)d_out);
}